// STChebConv_88167088652338
// MI455X (gfx1250) — compile-verified
//
#include <hip/hip_runtime.h>
#include <hip/hip_bf16.h>

typedef __attribute__((ext_vector_type(2))) float v2f;
typedef __attribute__((ext_vector_type(8))) float v8f;

#define FDIM 144
#define KCHEB 3

// ---------------------------------------------------------------------------
// Utility: zero a float buffer
// ---------------------------------------------------------------------------
__global__ void zero_f32(float* __restrict__ p, int n) {
    int i = blockIdx.x * blockDim.x + threadIdx.x;
    if (i < n) p[i] = 0.0f;
}

// ---------------------------------------------------------------------------
// Degree: deg[row[e]] += 1 for non-self-loop edges
// ---------------------------------------------------------------------------
__global__ void degree_kernel(const int* __restrict__ row, const int* __restrict__ col,
                              float* __restrict__ deg, int nE) {
    int e = blockIdx.x * blockDim.x + threadIdx.x;
    if (e >= nE) return;
    int r = row[e], c = col[e];
    if (r != c) atomicAdd(&deg[r], 1.0f);
}

// deg -> deg^{-1/2} in place (0 if deg == 0)
__global__ void dinv_kernel(float* __restrict__ deg, int n) {
    int i = blockIdx.x * blockDim.x + threadIdx.x;
    if (i >= n) return;
    float d = deg[i];
    deg[i] = (d > 0.0f) ? rsqrtf(d) : 0.0f;
}

// lap[e] = -dinv[row]*w*dinv[col], zero for self loops
__global__ void lap_kernel(const int* __restrict__ row, const int* __restrict__ col,
                           const float* __restrict__ w, const float* __restrict__ dinv,
                           float* __restrict__ lap, int nE) {
    int e = blockIdx.x * blockDim.x + threadIdx.x;
    if (e >= nE) return;
    int r = row[e], c = col[e];
    lap[e] = (r != c) ? (-dinv[r] * w[e] * dinv[c]) : 0.0f;
}

// ---------------------------------------------------------------------------
// SpMM scatter: dst[row[e], :] += lap[e] * src[col[e], :]
// One thread handles 4 contiguous features of one edge (float4 gather,
// 4x global_atomic_add_f32 scatter).
// ---------------------------------------------------------------------------
__global__ void spmm_scatter(const int* __restrict__ row, const int* __restrict__ col,
                             const float* __restrict__ lap, const float* __restrict__ src,
                             float* __restrict__ dst, int nE) {
    int idx = blockIdx.x * blockDim.x + threadIdx.x;
    int total = nE * (FDIM / 4);
    if (idx >= total) return;
    int e  = idx / (FDIM / 4);
    int f4 = (idx % (FDIM / 4)) * 4;
    float l = lap[e];
    if (l == 0.0f) return;
    int c = col[e];
    int r = row[e];
    const float4 v = *(const float4*)(&src[c * FDIM + f4]);
    float* o = &dst[r * FDIM + f4];
    atomicAdd(o + 0, l * v.x);
    atomicAdd(o + 1, l * v.y);
    atomicAdd(o + 2, l * v.z);
    atomicAdd(o + 3, l * v.w);
}

// ---------------------------------------------------------------------------
// Fused Chebyshev GEMM (fp32 WMMA):
//   out = x @ W0 + tx1 @ W1 + (2*s - x) @ W2 + bias
// One wave (32 threads) per 16x16 output tile. N=50000 and F=144 are exact
// multiples of 16 -> every tile is full, EXEC all-ones as WMMA requires.
//
// Fragment layouts per CDNA5 ISA 7.12.2:
//  A (16x4 f32): lanes 0-15 -> M=lane, {K, K+1}; lanes 16-31 -> M=lane-16, {K+2, K+3}
//  B (4x16 f32): lanes 0-15 -> N=lane, {K, K+1}; lanes 16-31 -> N=lane-16, {K+2, K+3}
//  C/D (16x16 f32, 8 VGPRs): VGPR v, lane l -> M = v + (l>=16 ? 8 : 0), N = l&15
// ---------------------------------------------------------------------------
__global__ __launch_bounds__(32) void cheb_gemm_wmma(
    const float* __restrict__ x,    // N x 144  (Tx0)
    const float* __restrict__ tx1,  // N x 144
    const float* __restrict__ s,    // N x 144  (spmm(Tx1); Tx2 = 2*s - x)
    const float* __restrict__ w,    // 3 x 144 x 144
    const float* __restrict__ bias, // 144
    float* __restrict__ out)        // N x 144
{
    const int rowBase = blockIdx.x * 16;
    const int colBase = blockIdx.y * 16;
    const int lane    = threadIdx.x;       // 0..31
    const int m       = lane & 15;
    const int khalf   = (lane >> 4) * 2;   // 0 or 2

    const int aRow = (rowBase + m) * FDIM;
    v8f acc = {};

    #pragma unroll
    for (int t = 0; t < KCHEB; ++t) {
        const float* wt = w + t * (FDIM * FDIM);
        for (int k = 0; k < FDIM; k += 4) {
            const int ka = k + khalf;
            v2f a;
            if (t == 0) {
                a = *(const v2f*)(&x[aRow + ka]);
            } else if (t == 1) {
                a = *(const v2f*)(&tx1[aRow + ka]);
            } else {
                v2f sv = *(const v2f*)(&s[aRow + ka]);
                v2f xv = *(const v2f*)(&x[aRow + ka]);
                a = 2.0f * sv - xv;
            }
            v2f b;
            b.x = wt[(ka + 0) * FDIM + colBase + m];
            b.y = wt[(ka + 1) * FDIM + colBase + m];
            acc = __builtin_amdgcn_wmma_f32_16x16x4_f32(
                /*neg_a=*/false, a, /*neg_b=*/false, b,
                /*c_mod=*/(short)0, acc, /*reuse_a=*/false, /*reuse_b=*/false);
        }
    }

    const float bs  = bias[colBase + m];
    const int rOff  = (lane >> 4) * 8;
    #pragma unroll
    for (int v = 0; v < 8; ++v) {
        out[(rowBase + rOff + v) * FDIM + colBase + m] = acc[v] + bs;
    }
}

// ---------------------------------------------------------------------------
// Host-side launcher
// ---------------------------------------------------------------------------
extern "C" void kernel_launch(void* const* d_in, const int* in_sizes, int n_in,
                              void* d_out, int out_size, void* d_ws, size_t ws_size,
                              hipStream_t stream) {
    const float* x      = (const float*)d_in[0];   // N*144
    const int*   eidx   = (const int*)d_in[1];     // 2*E
    const float* ew     = (const float*)d_in[2];   // E
    const float* weight = (const float*)d_in[3];   // 3*144*144
    const float* bias   = (const float*)d_in[4];   // 144
    float*       out    = (float*)d_out;

    const int E = in_sizes[2];
    const int N = in_sizes[0] / FDIM;
    const int NF = N * FDIM;

    const int* row = eidx;
    const int* col = eidx + E;

    // Workspace layout (floats): [tx1: NF][s: NF][lap: E][dinv: N]
    float* ws   = (float*)d_ws;
    float* tx1  = ws;
    float* sbuf = ws + NF;
    float* lap  = ws + 2 * NF;
    float* dinv = ws + 2 * NF + E;

    const int TB = 256;

    // 1. zero accumulators (tx1 + s are contiguous) and degree buffer
    zero_f32<<<(2 * NF + TB - 1) / TB, TB, 0, stream>>>(tx1, 2 * NF);
    zero_f32<<<(N + TB - 1) / TB, TB, 0, stream>>>(dinv, N);

    // 2. degree, 3. dinv, 4. normalized Laplacian edge weights
    degree_kernel<<<(E + TB - 1) / TB, TB, 0, stream>>>(row, col, dinv, E);
    dinv_kernel<<<(N + TB - 1) / TB, TB, 0, stream>>>(dinv, N);
    lap_kernel<<<(E + TB - 1) / TB, TB, 0, stream>>>(row, col, ew, dinv, lap, E);

    // 5. Tx1 = spmm(x),  6. S = spmm(Tx1)   (Tx2 = 2S - x fused into GEMM)
    {
        long long total = (long long)E * (FDIM / 4);
        int blocks = (int)((total + TB - 1) / TB);
        spmm_scatter<<<blocks, TB, 0, stream>>>(row, col, lap, x, tx1, E);
        spmm_scatter<<<blocks, TB, 0, stream>>>(row, col, lap, tx1, sbuf, E);
    }

    // 7. Fused triple GEMM + bias on fp32 WMMA; all tiles full (N%16==0, 144%16==0)
    dim3 grid(N / 16, FDIM / 16);
    cheb_gemm_wmma<<<grid, 32, 0, stream>>>(x, tx1, sbuf, weight, bias, out);
}